// DGGraphConv_24781961298372
// MI455X (gfx1250) — compile-verified
//
#include <hip/hip_runtime.h>

// D = A*B+C via V_WMMA_F32_16X16X4_F32 (wave32, gfx1250)
typedef __attribute__((ext_vector_type(2))) float v2f;
typedef __attribute__((ext_vector_type(8))) float v8f;

#define IN_SIZE  256
#define OUT_SIZE 256
#define LDA 268   // padded LDS row stride for A (floats): 12*j mod 64 distinct, halves never collide
#define LDB 72    // padded LDS row stride for B (floats): half-waves land 16 banks apart

// ---------------------------------------------------------------------------
// Kernel 1: supp = X @ W using fp32 WMMA. Block = 128 threads = 4 waves.
// Block computes a 16(M) x 64(N) tile; wave w covers columns [w*16, w*16+16).
// K = 256 split into FOUR independent accumulator chains (16 WMMAs each) so
// each SIMD32 sees ~12 concurrent V_WMMA chains -> matrix pipe stays full
// despite the C-operand RAW dependency within each chain.
// ---------------------------------------------------------------------------
__global__ __launch_bounds__(128)
void gcn_gemm_wmma(const float* __restrict__ X, const float* __restrict__ W,
                   float* __restrict__ S, int M) {
    __shared__ float lA[16 * LDA];        // 16 x 256 A tile (padded)
    __shared__ float lB[IN_SIZE * LDB];   // 256 x 64 B slab (padded)

    const int t    = threadIdx.x;
    const int lane = t & 31;
    const int wv   = t >> 5;        // wave id 0..3
    const int lm   = lane & 15;     // M (for A) / N (for B) index within 16
    const int half = lane >> 4;     // 0: K={0,1}  1: K={2,3}  (ISA 16x4 f32 A layout)

    const int row0 = blockIdx.x * 16;
    const int colb = blockIdx.y * 64;
    const int wcol = wv * 16;

    // Stage A tile (16 x 256 floats) -> LDS, float4 granularity, coalesced.
    #pragma unroll
    for (int i = 0; i < 8; ++i) {
        int idx4 = t + i * 128;
        int r    = idx4 >> 6;       // 64 float4 per row
        int c4   = idx4 & 63;
        float4 v = make_float4(0.f, 0.f, 0.f, 0.f);
        if (row0 + r < M)
            v = *(const float4*)(X + (size_t)(row0 + r) * IN_SIZE + c4 * 4);
        *(float4*)(&lA[r * LDA + c4 * 4]) = v;  // LDA*4 bytes % 16 == 0 -> aligned
    }
    // Stage B slab (256 x 64 floats) -> LDS.
    #pragma unroll
    for (int i = 0; i < 32; ++i) {
        int idx4 = t + i * 128;
        int r    = idx4 >> 4;       // 16 float4 per row
        int c4   = idx4 & 15;
        float4 v = *(const float4*)(W + (size_t)r * OUT_SIZE + colb + c4 * 4);
        *(float4*)(&lB[r * LDB + c4 * 4]) = v;
    }
    __syncthreads();

    const int aBase = lm * LDA + 2 * half;          // A: row lm, K-offset of this half-wave
    const int bBase = 2 * half * LDB + wcol + lm;   // B: K-row of this half-wave, col lm

    v8f acc0 = {0.f,0.f,0.f,0.f,0.f,0.f,0.f,0.f};
    v8f acc1 = acc0, acc2 = acc0, acc3 = acc0;

    #pragma unroll
    for (int k = 0; k < 64; k += 4) {
        // chain 0: K segment [0,64)
        {
            v2f a = *(const v2f*)(&lA[aBase + k]);
            v2f b; b.x = lB[bBase + k * LDB]; b.y = lB[bBase + (k + 1) * LDB];
            acc0 = __builtin_amdgcn_wmma_f32_16x16x4_f32(false, a, false, b, (short)0, acc0, false, false);
        }
        // chain 1: K segment [64,128)
        {
            v2f a = *(const v2f*)(&lA[aBase + k + 64]);
            v2f b; b.x = lB[bBase + (k + 64) * LDB]; b.y = lB[bBase + (k + 65) * LDB];
            acc1 = __builtin_amdgcn_wmma_f32_16x16x4_f32(false, a, false, b, (short)0, acc1, false, false);
        }
        // chain 2: K segment [128,192)
        {
            v2f a = *(const v2f*)(&lA[aBase + k + 128]);
            v2f b; b.x = lB[bBase + (k + 128) * LDB]; b.y = lB[bBase + (k + 129) * LDB];
            acc2 = __builtin_amdgcn_wmma_f32_16x16x4_f32(false, a, false, b, (short)0, acc2, false, false);
        }
        // chain 3: K segment [192,256)
        {
            v2f a = *(const v2f*)(&lA[aBase + k + 192]);
            v2f b; b.x = lB[bBase + (k + 192) * LDB]; b.y = lB[bBase + (k + 193) * LDB];
            acc3 = __builtin_amdgcn_wmma_f32_16x16x4_f32(false, a, false, b, (short)0, acc3, false, false);
        }
    }

    v8f acc = (acc0 + acc1) + (acc2 + acc3);   // merge the 4 partial K-sums

    // C/D 16x16 f32 layout: VGPR v, lane -> (M = v + 8*half, N = lm)
    #pragma unroll
    for (int v = 0; v < 8; ++v) {
        int r = row0 + v + 8 * half;
        if (r < M)
            S[(size_t)r * OUT_SIZE + colb + wcol + lm] = acc[v];
    }
}

// ---------------------------------------------------------------------------
// Kernel 2: out[i][j] = bias[j]  (so the scatter can accumulate on top)
// ---------------------------------------------------------------------------
__global__ __launch_bounds__(256)
void gcn_init_bias(float* __restrict__ out, const float* __restrict__ bias,
                   long long total) {
    long long i = (long long)blockIdx.x * blockDim.x + threadIdx.x;
    if (i < total) out[i] = bias[i & (OUT_SIZE - 1)];
}

// ---------------------------------------------------------------------------
// Kernel 3: one wave32 per edge. Each lane handles 8 columns:
//   out[dst][c] += supp[src][c] * w   via global_atomic_add_f32 (L2-resident)
// supp (102 MB) and out (102 MB) both fit in the 192 MB global L2.
// ---------------------------------------------------------------------------
__global__ __launch_bounds__(256)
void gcn_edge_scatter(const float* __restrict__ S, const float* __restrict__ ew,
                      const int* __restrict__ esrc, const int* __restrict__ edst,
                      float* __restrict__ out, int E) {
    int e    = (int)(((long long)blockIdx.x * blockDim.x + threadIdx.x) >> 5);
    int lane = threadIdx.x & 31;
    if (e >= E) return;
    int   s = esrc[e];
    int   d = edst[e];
    float w = ew[e];
    const float* sp = S   + (size_t)s * OUT_SIZE;
    float*       op = out + (size_t)d * OUT_SIZE;
    #pragma unroll
    for (int i = 0; i < 8; ++i) {
        int c = lane + i * 32;                  // coalesced 128B per access
        atomicAdd(&op[c], sp[c] * w);
    }
}

extern "C" void kernel_launch(void* const* d_in, const int* in_sizes, int n_in,
                              void* d_out, int out_size, void* d_ws, size_t ws_size,
                              hipStream_t stream) {
    const float* x    = (const float*)d_in[0];   // [N, 256]
    const float* w    = (const float*)d_in[1];   // [256, 256]
    const float* bias = (const float*)d_in[2];   // [1, 256]
    const float* ew   = (const float*)d_in[3];   // [E]
    const int*   esrc = (const int*)d_in[4];     // [E]
    const int*   edst = (const int*)d_in[5];     // [E]
    float* out  = (float*)d_out;                 // [N, 256]
    float* supp = (float*)d_ws;                  // [N, 256] scratch (102.4 MB)

    const int M = in_sizes[0] / IN_SIZE;
    const int E = in_sizes[3];

    // 1) supp = x @ W  (WMMA f32, 4 independent accumulator chains per wave)
    dim3 ggrid((M + 15) / 16, OUT_SIZE / 64);
    gcn_gemm_wmma<<<ggrid, 128, 0, stream>>>(x, w, supp, M);

    // 2) out = broadcast(bias)
    long long total = (long long)M * OUT_SIZE;
    int iblocks = (int)((total + 255) / 256);
    gcn_init_bias<<<iblocks, 256, 0, stream>>>(out, bias, total);

    // 3) out[dst] += supp[src] * edge_weight   (atomic scatter, 8 edges/block)
    int sblocks = (E + 7) / 8;
    gcn_edge_scatter<<<sblocks, 256, 0, stream>>>(supp, ew, esrc, edst, out, E);
}